// GraphNorm_81784767250589
// MI455X (gfx1250) — compile-verified
//
#include <hip/hip_runtime.h>
#include <hip/hip_bf16.h>

typedef __attribute__((ext_vector_type(2))) float v2f;
typedef __attribute__((ext_vector_type(4))) float v4f;
typedef __attribute__((ext_vector_type(8))) float v8f;

#define EPSV 1e-5f
#define FDIM 256   // fixed feature dimension for this problem

// ---------------------------------------------------------------------------
// Kernel 1: offs[g] = lower_bound(batch, g) for g in [0, G]; batch is sorted.
// ---------------------------------------------------------------------------
__global__ void GraphNorm_offsets_kernel(const int* __restrict__ batch,
                                         const int* __restrict__ gptr,
                                         int N, int* __restrict__ offs) {
    const int G = *gptr;
    const int tid = blockIdx.x * blockDim.x + threadIdx.x;
    if (tid > G) return;
    int lo = 0, hi = N;
    while (lo < hi) {
        int mid = (lo + hi) >> 1;
        if (batch[mid] < tid) lo = mid + 1; else hi = mid;
    }
    offs[tid] = lo;
}

// ---------------------------------------------------------------------------
// Kernel 2: one 256-thread block per group (grid-stride over groups).
// Pass 1: column sum / sumsq via V_WMMA_F32_16X16X4_F32 with ones-A:
//         D = ones(16x4) * B(4x16) + C  => every row of D is the column sum.
//         All 4 loads of a chunk share ONE index VGPR + byte immediates
//         {0, 64, 1024, 1088}; loop update is a single idx += 1024.
// Pass 2: float4 per lane, out = fma(x, a, b), non-temporal b128 streamed.
// Requires blockDim.x == 256 (8 waves * 32 cols), F == 256.
// ---------------------------------------------------------------------------
__global__ void GraphNorm_main_kernel(const float* __restrict__ x,
                                      const float* __restrict__ weight,
                                      const float* __restrict__ bias,
                                      const int* __restrict__ gptr,
                                      const int* __restrict__ offs,
                                      float* __restrict__ out) {
    const int G = *gptr;
    __shared__ __align__(16) float s_a[FDIM];
    __shared__ __align__(16) float s_b[FDIM];

    const int tid  = threadIdx.x;
    const int wave = tid >> 5;
    const int lane = tid & 31;

    // B-matrix (4x16 f32) lane layout per ISA 7.12.2:
    // lanes 0-15: VGPR0=K0, VGPR1=K1 ; lanes 16-31: VGPR0=K2, VGPR1=K3
    const int kbase = (lane < 16) ? 0 : 2;
    const int nIdx  = lane & 15;
    const int col0  = (wave << 5) + nIdx;   // tile 0 column
    const int col1  = col0 + 16;            // tile 1 column

    // pass-2 mapping: thread handles float4 column group t4, row offset rOff
    const int t4   = tid & 63;
    const int rOff = tid >> 6;

    const v2f ones = {1.0f, 1.0f};

    for (int g = blockIdx.x; g < G; g += gridDim.x) {
        const int start = offs[g];
        const int end   = offs[g + 1];
        const int cnt   = end - start;

        v8f accS0 = {}; v8f accQ0 = {};
        v8f accS1 = {}; v8f accQ1 = {};

        // ---- pass 1 main loop: full 4-row chunks, unconditional loads ----
        int r   = start;
        int idx = (start + kbase) * FDIM + col0;   // element index
#pragma unroll 2
        for (; r < end - 3; r += 4, idx += 4 * FDIM) {
            // prefetch 32 rows ahead into near caches (speculative; OOB ok)
            __builtin_prefetch(&x[idx + 32 * FDIM], 0, 3);

            const float b00 = x[idx];               // row r0,   col0
            const float b01 = x[idx + FDIM];        // row r0+1, col0
            const float b10 = x[idx + 16];          // row r0,   col1
            const float b11 = x[idx + FDIM + 16];   // row r0+1, col1

            const v2f B0 = {b00, b01};
            const v2f B1 = {b10, b11};
            const v2f Q0 = {b00 * b00, b01 * b01};
            const v2f Q1 = {b10 * b10, b11 * b11};

            accS0 = __builtin_amdgcn_wmma_f32_16x16x4_f32(
                false, ones, false, B0, (short)0, accS0, false, false);
            accQ0 = __builtin_amdgcn_wmma_f32_16x16x4_f32(
                false, ones, false, Q0, (short)0, accQ0, false, false);
            accS1 = __builtin_amdgcn_wmma_f32_16x16x4_f32(
                false, ones, false, B1, (short)0, accS1, false, false);
            accQ1 = __builtin_amdgcn_wmma_f32_16x16x4_f32(
                false, ones, false, Q1, (short)0, accQ1, false, false);
        }

        // ---- tail: one guarded chunk (cnt % 4 rows), outside hot loop ----
        if (r < end) {
            const int r0 = r + kbase;
            const int r1 = r0 + 1;

            const float b00 = (r0 < end) ? x[idx] : 0.0f;
            const float b01 = (r1 < end) ? x[idx + FDIM] : 0.0f;
            const float b10 = (r0 < end) ? x[idx + 16] : 0.0f;
            const float b11 = (r1 < end) ? x[idx + FDIM + 16] : 0.0f;

            const v2f B0 = {b00, b01};
            const v2f B1 = {b10, b11};
            const v2f Q0 = {b00 * b00, b01 * b01};
            const v2f Q1 = {b10 * b10, b11 * b11};

            accS0 = __builtin_amdgcn_wmma_f32_16x16x4_f32(
                false, ones, false, B0, (short)0, accS0, false, false);
            accQ0 = __builtin_amdgcn_wmma_f32_16x16x4_f32(
                false, ones, false, Q0, (short)0, accQ0, false, false);
            accS1 = __builtin_amdgcn_wmma_f32_16x16x4_f32(
                false, ones, false, B1, (short)0, accS1, false, false);
            accQ1 = __builtin_amdgcn_wmma_f32_16x16x4_f32(
                false, ones, false, Q1, (short)0, accQ1, false, false);
        }

        // Row M=0 of D lives in accumulator element 0, lanes 0-15 (N=0..15).
        if (lane < 16) {
            const float cs   = (float)cnt;
            const float cdiv = fmaxf(cs, 1.0f);
            const float ddiv = fmaxf(cs - 1.0f, 1.0f);

            {   // tile 0
                const float sum  = accS0[0];
                const float sq   = accQ0[0];
                const float mean = sum / cdiv;
                const float var  = (sq - mean * sum) / ddiv;
                const float stdv = sqrtf(fmaxf(var, 0.0f)) + EPSV;
                const float a    = weight[col0] / stdv;
                s_a[col0] = a;
                s_b[col0] = bias[col0] - mean * a;
            }
            {   // tile 1
                const float sum  = accS1[0];
                const float sq   = accQ1[0];
                const float mean = sum / cdiv;
                const float var  = (sq - mean * sum) / ddiv;
                const float stdv = sqrtf(fmaxf(var, 0.0f)) + EPSV;
                const float a    = weight[col1] / stdv;
                s_a[col1] = a;
                s_b[col1] = bias[col1] - mean * a;
            }
        }
        __syncthreads();

        // ---- pass 2: b128 per lane; x re-read hits L2, output streamed ----
        const v4f av = ((const v4f*)s_a)[t4];   // ds_load_b128
        const v4f bv = ((const v4f*)s_b)[t4];

        int pidx = (start + rOff) * FDIM + (t4 << 2);
        for (int rr = start + rOff; rr < end; rr += 4, pidx += 4 * FDIM) {
            const v4f v = __builtin_nontemporal_load((const v4f*)&x[pidx]);
            v4f o;
            o.x = fmaf(v.x, av.x, bv.x);
            o.y = fmaf(v.y, av.y, bv.y);
            o.z = fmaf(v.z, av.z, bv.z);
            o.w = fmaf(v.w, av.w, bv.w);
            __builtin_nontemporal_store(o, (v4f*)&out[pidx]);
        }
        __syncthreads();
    }
}

// ---------------------------------------------------------------------------
// Host-side launcher
// Inputs (setup_inputs order): x [N*F f32], batch [N int], num_graphs [1 int],
//                              weight [F f32], bias [F f32]
// ---------------------------------------------------------------------------
extern "C" void kernel_launch(void* const* d_in, const int* in_sizes, int n_in,
                              void* d_out, int out_size, void* d_ws, size_t ws_size,
                              hipStream_t stream) {
    const float* x      = (const float*)d_in[0];
    const int*   batch  = (const int*)d_in[1];
    const int*   gptr   = (const int*)d_in[2];
    const float* weight = (const float*)d_in[3];
    const float* bias   = (const float*)d_in[4];
    float*       outp   = (float*)d_out;

    const int N = in_sizes[1];

    int* offs = (int*)d_ws;      // (G+1) ints, G <= 4095 supported

    GraphNorm_offsets_kernel<<<16, 256, 0, stream>>>(batch, gptr, N, offs);
    GraphNorm_main_kernel<<<512, 256, 0, stream>>>(x, weight, bias, gptr, offs,
                                                   outp);
}